// TensorProduct_47648367182206
// MI455X (gfx1250) — compile-verified
//
#include <hip/hip_runtime.h>

typedef __attribute__((ext_vector_type(2))) float v2f;
typedef __attribute__((ext_vector_type(8))) float v8f;

// Clebsch-Gordan change-of-basis matrices, computed on device each launch.
__device__ float g_U25[625]; // pl=4 (band 3), row-major [M][N], M=(i,j) flat, N=L^2+k
__device__ float g_U9[81];   // pl=2 (bands 1,2)

__device__ __forceinline__ v8f wmma4(v2f a, v2f b, v8f c) {
  // D(16x16,f32) = A(16x4,f32) * B(4x16,f32) + C
  return __builtin_amdgcn_wmma_f32_16x16x4_f32(false, a, false, b, (short)0, c,
                                               false, false);
}

// ---------------- U-matrix construction (Racah formula, fp64) ----------------
__device__ __forceinline__ int imax(int a, int b) { return a > b ? a : b; }
__device__ __forceinline__ int imin(int a, int b) { return a < b ? a : b; }

__device__ double dfac(int x) {
  double r = 1.0;
  for (int i = 2; i <= x; ++i) r *= (double)i;
  return r;
}

__device__ double cgc(int j1, int m1, int j2, int m2, int J, int M) {
  if (m1 + m2 != M) return 0.0;
  double pref = (2.0 * J + 1.0) * dfac(j1 + j2 - J) * dfac(j1 - j2 + J) *
                dfac(-j1 + j2 + J) / dfac(j1 + j2 + J + 1);
  pref *= dfac(J + M) * dfac(J - M) * dfac(j1 - m1) * dfac(j1 + m1) *
          dfac(j2 - m2) * dfac(j2 + m2);
  int kmin = imax(0, imax(j2 - J - m1, j1 + m2 - J));
  int kmax = imin(j1 + j2 - J, imin(j1 - m1, j2 + m2));
  double s = 0.0;
  for (int k = kmin; k <= kmax; ++k) {
    double d = dfac(k) * dfac(j1 + j2 - J - k) * dfac(j1 - m1 - k) *
               dfac(j2 + m2 - k) * dfac(J - j2 + m1 + k) * dfac(J - j1 - m2 + k);
    s += ((k & 1) ? -1.0 : 1.0) / d;
  }
  return sqrt(pref) * s;
}

// real->complex spherical harmonic basis matrix element R_L[r][c]
__device__ void r2c(int L, int r, int c, double &re, double &im) {
  const double isq2 = 0.70710678118654752440;
  int m = c - L;
  re = 0.0; im = 0.0;
  double s = (m & 1) ? -1.0 : 1.0; // (-1)^m
  if (m < 0) {
    if (r == 2 * L - c)      im = isq2 * s;
    else if (r == c)         im = -isq2;
  } else if (m == 0) {
    if (r == L)              re = 1.0;
  } else {
    if (r == 2 * L - c)      re = isq2;
    else if (r == c)         re = isq2 * s;
  }
}

// real CG coefficient cg_real(h,h,L)[a,b,k]
__device__ float u_entry(int h, int a, int b, int L, int k) {
  double re = 0.0, im = 0.0;
  for (int m1 = 0; m1 <= 2 * h; ++m1) {
    double r1r, r1i; r2c(h, m1, a, r1r, r1i);
    if (r1r == 0.0 && r1i == 0.0) continue;
    for (int m2 = 0; m2 <= 2 * h; ++m2) {
      double r2r, r2i; r2c(h, m2, b, r2r, r2i);
      if (r2r == 0.0 && r2i == 0.0) continue;
      int Mi = m1 + m2 - 2 * h + L; // M index in 0..2L
      if (Mi < 0 || Mi > 2 * L) continue;
      double cc = cgc(h, m1 - h, h, m2 - h, L, Mi - L);
      if (cc == 0.0) continue;
      double r3r, r3i; r2c(L, Mi, k, r3r, r3i);
      r3i = -r3i; // conjugate
      double pr = r1r * r2r - r1i * r2i;
      double pi = r1r * r2i + r1i * r2r;
      double qr = pr * r3r - pi * r3i;
      double qi = pr * r3i + pi * r3r;
      re += qr * cc; im += qi * cc;
    }
  }
  return (float)((L & 1) ? im : re); // parity (2h+L)%2 == L%2
}

__global__ __launch_bounds__(256) void init_U_kernel() {
  int t = blockIdx.x * blockDim.x + threadIdx.x;
  if (t < 625) {
    int row = t / 25, c = t % 25;
    int a = row / 5, b = row % 5;
    int L = (c >= 1) + (c >= 4) + (c >= 9) + (c >= 16);
    int k = c - L * L;
    g_U25[t] = u_entry(2, a, b, L, k);
  } else if (t < 625 + 81) {
    int q = t - 625;
    int row = q / 9, c = q % 9;
    int a = row / 3, b = row % 3;
    int L = (c >= 1) + (c >= 4);
    int k = c - L * L;
    g_U9[q] = u_entry(1, a, b, L, k);
  }
}

// ---------------- safe element accessors ----------------
__device__ __forceinline__ float u25(int r, int c) {
  bool ok = (r >= 0) && (r < 25) && (c >= 0) && (c < 25);
  int i = ok ? (r * 25 + c) : 0;
  float v = g_U25[i];
  return ok ? v : 0.0f;
}
__device__ __forceinline__ float u9(int r, int c) {
  bool ok = (r >= 0) && (r < 9) && (c >= 0) && (c < 9);
  int i = ok ? (r * 9 + c) : 0;
  float v = g_U9[i];
  return ok ? v : 0.0f;
}

// stacked[r, col] for one sample of one band; rows >= realN are the zero pad
__device__ __forceinline__ float xval(const float* __restrict__ f0,
                                      const float* __restrict__ f1,
                                      const float* __restrict__ f2,
                                      const float* __restrict__ f3,
                                      int samp, int r, int ch0, int col,
                                      int realN) {
  bool ok = (r < realN);
  int rc = ok ? r : 0;
  int L = (rc >= 1) + (rc >= 4) + (rc >= 9); // 0..3
  int k = rc - L * L;
  int K = 64 * (4 - L); // 256,192,128,64
  const float* f = (L == 0) ? f0 : ((L == 1) ? f1 : ((L == 2) ? f2 : f3));
  float v = f[((size_t)samp * (2 * L + 1) + k) * (size_t)K + (size_t)(ch0 + col)];
  return ok ? v : 0.0f;
}

// ---------------- band 3: pl=4, N=25, real K=16 ----------------
__global__ __launch_bounds__(256) void band3_kernel(
    const float* __restrict__ f10, const float* __restrict__ f11,
    const float* __restrict__ f12, const float* __restrict__ f13,
    const float* __restrict__ f20, const float* __restrict__ f21,
    const float* __restrict__ f22, const float* __restrict__ f23,
    float* __restrict__ out, int n) {
  __shared__ float sm[8 * 2 * 544]; // per-wave su1/su2, 32 rows x stride 17
  const int lane = threadIdx.x & 31;
  const int wid = threadIdx.x >> 5;
  const int lm = lane & 15;  // WMMA M (A-row) / N (B,D column)
  const int hl = lane >> 4;  // lane half selects K pair {0,1} vs {2,3}
  int task = blockIdx.x * 8 + wid;
  int samp = task >> 2;
  if (samp >= n) return; // wave-uniform
  int t = task & 3;
  int col = t * 16 + lm; // channel 0..63 within the band slice
  float* su1 = sm + wid * 1088;
  float* su2 = su1 + 544;

  // uncouple: u = U25[:, :16] @ X[:16]  (rows 16..24 of stacked are zero pad)
  v8f u1a = {}, u1b = {}, u2a = {}, u2b = {};
#pragma unroll
  for (int ch = 0; ch < 4; ++ch) {
    int r0 = ch * 4 + 2 * hl; // this lane's K rows: r0, r0+1 (<=15)
    v2f b1, b2, aa, ab;
    b1.x = xval(f10, f11, f12, f13, samp, r0, 0, col, 16);
    b1.y = xval(f10, f11, f12, f13, samp, r0 + 1, 0, col, 16);
    b2.x = xval(f20, f21, f22, f23, samp, r0, 0, col, 16);
    b2.y = xval(f20, f21, f22, f23, samp, r0 + 1, 0, col, 16);
    aa.x = u25(lm, r0);       aa.y = u25(lm, r0 + 1);        // M-tile 0..15
    ab.x = u25(16 + lm, r0);  ab.y = u25(16 + lm, r0 + 1);   // M-tile 16..31
    u1a = wmma4(aa, b1, u1a);
    u1b = wmma4(ab, b1, u1b);
    u2a = wmma4(aa, b2, u2a);
    u2b = wmma4(ab, b2, u2b);
  }

  // D layout -> LDS (row-major, stride 17): lane hl=0 holds rows v/16+v, hl=1 rows 8+v/24+v
#pragma unroll
  for (int v = 0; v < 8; ++v) {
    int ra = hl * 8 + v;
    su1[ra * 17 + lm] = u1a[v];
    su1[(16 + ra) * 17 + lm] = u1b[v];
    su2[ra * 17 + lm] = u2a[v];
    su2[(16 + ra) * 17 + lm] = u2b[v];
  }
  __builtin_amdgcn_wave_barrier();

  // per-column 5x5 matmul: comb = mat(u1) @ mat(u2); write into su1
  float x1[25], x2[25];
#pragma unroll
  for (int r = 0; r < 25; ++r) { x1[r] = su1[r * 17 + lm]; x2[r] = su2[r * 17 + lm]; }
  __builtin_amdgcn_wave_barrier();
#pragma unroll
  for (int i = 0; i < 5; ++i) {
#pragma unroll
    for (int k = 0; k < 5; ++k) {
      float s = 0.0f;
#pragma unroll
      for (int j = 0; j < 5; ++j) s = fmaf(x1[i * 5 + j], x2[j * 5 + k], s);
      su1[(i * 5 + k) * 17 + lm] = s;
    }
  }
  __builtin_amdgcn_wave_barrier();

  // couple: coup[M=0..15] = U25^T[:16, :25] @ comb  (L=4 rows 16..24 discarded)
  v8f acc = {};
#pragma unroll
  for (int ch = 0; ch < 7; ++ch) {
    int r0 = ch * 4 + 2 * hl;
    v2f a, b;
    a.x = u25(r0, lm);       // Ut[M=lm, K=r0] = U[r0, lm]
    a.y = u25(r0 + 1, lm);
    b.x = (r0 < 25) ? su1[r0 * 17 + lm] : 0.0f;
    b.y = (r0 + 1 < 25) ? su1[(r0 + 1) * 17 + lm] : 0.0f;
    acc = wmma4(a, b, acc);
  }

  // scatter rows 0..15 -> out_L[n,k, 64*(3-L)+col]
  size_t OB1 = (size_t)n * 256;
  size_t OB2 = OB1 + (size_t)n * 576;
  size_t OB3 = OB2 + (size_t)n * 640;
#pragma unroll
  for (int v = 0; v < 8; ++v) {
    int m = hl * 8 + v; // 0..15
    int L = (m >= 1) + (m >= 4) + (m >= 9);
    int k = m - L * L;
    int W = 64 * (4 - L);
    size_t base = (L == 0) ? 0 : ((L == 1) ? OB1 : ((L == 2) ? OB2 : OB3));
    out[base + ((size_t)samp * (2 * L + 1) + k) * (size_t)W +
        (size_t)(64 * (3 - L) + col)] = acc[v];
  }
}

// ---------------- bands 1,2: pl=2, N=9 ----------------
__global__ __launch_bounds__(256) void band12_kernel(
    const float* __restrict__ f10, const float* __restrict__ f11,
    const float* __restrict__ f12, const float* __restrict__ f13,
    const float* __restrict__ f20, const float* __restrict__ f21,
    const float* __restrict__ f22, const float* __restrict__ f23,
    float* __restrict__ out, int n, int band) {
  __shared__ float sm[8 * 2 * 272];
  const int lane = threadIdx.x & 31;
  const int wid = threadIdx.x >> 5;
  const int lm = lane & 15;
  const int hl = lane >> 4;
  int task = blockIdx.x * 8 + wid;
  int samp = task >> 2;
  if (samp >= n) return;
  int t = task & 3;
  int col = t * 16 + lm;
  int ch0 = (band == 1) ? 128 : 64;
  int realN = (band == 1) ? 4 : 9;   // real stacked rows (band1 pads L=2 block)
  int kcu = (band == 1) ? 1 : 3;     // K chunks covering realN
  float* su1 = sm + wid * 544;
  float* su2 = su1 + 272;

  v8f u1 = {}, u2 = {};
  for (int ch = 0; ch < kcu; ++ch) { // uniform trip count
    int r0 = ch * 4 + 2 * hl;
    v2f b1, b2, aa;
    b1.x = xval(f10, f11, f12, f13, samp, r0, ch0, col, realN);
    b1.y = xval(f10, f11, f12, f13, samp, r0 + 1, ch0, col, realN);
    b2.x = xval(f20, f21, f22, f23, samp, r0, ch0, col, realN);
    b2.y = xval(f20, f21, f22, f23, samp, r0 + 1, ch0, col, realN);
    aa.x = u9(lm, r0);
    aa.y = u9(lm, r0 + 1);
    u1 = wmma4(aa, b1, u1);
    u2 = wmma4(aa, b2, u2);
  }

#pragma unroll
  for (int v = 0; v < 8; ++v) {
    int ra = hl * 8 + v; // rows 9..15 are exact zeros (A guarded)
    su1[ra * 17 + lm] = u1[v];
    su2[ra * 17 + lm] = u2[v];
  }
  __builtin_amdgcn_wave_barrier();

  float x1[9], x2[9];
#pragma unroll
  for (int r = 0; r < 9; ++r) { x1[r] = su1[r * 17 + lm]; x2[r] = su2[r * 17 + lm]; }
  __builtin_amdgcn_wave_barrier();
#pragma unroll
  for (int i = 0; i < 3; ++i) {
#pragma unroll
    for (int k = 0; k < 3; ++k) {
      float s = 0.0f;
#pragma unroll
      for (int j = 0; j < 3; ++j) s = fmaf(x1[i * 3 + j], x2[j * 3 + k], s);
      su1[(i * 3 + k) * 17 + lm] = s;
    }
  }
  __builtin_amdgcn_wave_barrier();

  v8f acc = {};
#pragma unroll
  for (int ch = 0; ch < 3; ++ch) {
    int r0 = ch * 4 + 2 * hl;
    v2f a, b;
    a.x = u9(r0, lm);
    a.y = u9(r0 + 1, lm);
    b.x = (r0 < 16) ? su1[r0 * 17 + lm] : 0.0f;       // rows 9..15 are zeros
    b.y = (r0 + 1 < 16) ? su1[(r0 + 1) * 17 + lm] : 0.0f;
    acc = wmma4(a, b, acc);
  }

  int rowsOut = (band == 1) ? 4 : 9; // band1 keeps L0,L1; band2 keeps L0..L2
  size_t OB1 = (size_t)n * 256;
  size_t OB2 = OB1 + (size_t)n * 576;
#pragma unroll
  for (int v = 0; v < 8; ++v) {
    int m = hl * 8 + v;
    if (m < rowsOut) {
      int L = (m >= 1) + (m >= 4);
      int k = m - L * L;
      int W = 64 * (4 - L);
      size_t base = (L == 0) ? 0 : ((L == 1) ? OB1 : OB2);
      out[base + ((size_t)samp * (2 * L + 1) + k) * (size_t)W +
          (size_t)(64 * (band - L) + col)] = acc[v];
    }
  }
}

// ---------------- band 0: scalar product ----------------
__global__ void band0_kernel(const float* __restrict__ a,
                             const float* __restrict__ b,
                             float* __restrict__ out, int n) {
  int i = blockIdx.x * blockDim.x + threadIdx.x;
  if (i >= n * 64) return;
  int s = i >> 6, c = i & 63;
  size_t p = (size_t)s * 256;
  out[p + c] = a[p + 192 + c] * b[p + 192 + c];
}

extern "C" void kernel_launch(void* const* d_in, const int* in_sizes, int n_in,
                              void* d_out, int out_size, void* d_ws,
                              size_t ws_size, hipStream_t stream) {
  (void)n_in; (void)out_size; (void)d_ws; (void)ws_size;
  const float* f10 = (const float*)d_in[0];
  const float* f11 = (const float*)d_in[1];
  const float* f12 = (const float*)d_in[2];
  const float* f13 = (const float*)d_in[3];
  const float* f20 = (const float*)d_in[4];
  const float* f21 = (const float*)d_in[5];
  const float* f22 = (const float*)d_in[6];
  const float* f23 = (const float*)d_in[7];
  float* out = (float*)d_out;
  int n = in_sizes[0] / 256; // 32768

  init_U_kernel<<<3, 256, 0, stream>>>();
  band0_kernel<<<(n * 64 + 255) / 256, 256, 0, stream>>>(f10, f20, out, n);
  int blocks = (n * 4 + 7) / 8; // 8 wave-tasks per 256-thread block
  band12_kernel<<<blocks, 256, 0, stream>>>(f10, f11, f12, f13, f20, f21, f22,
                                            f23, out, n, 1);
  band12_kernel<<<blocks, 256, 0, stream>>>(f10, f11, f12, f13, f20, f21, f22,
                                            f23, out, n, 2);
  band3_kernel<<<blocks, 256, 0, stream>>>(f10, f11, f12, f13, f20, f21, f22,
                                           f23, out, n);
}